// Statistical_Proxy_Anchor_12163347383252
// MI455X (gfx1250) — compile-verified
//
#include <hip/hip_runtime.h>
#include <cmath>

typedef __attribute__((ext_vector_type(2))) float v2f;
typedef __attribute__((ext_vector_type(8))) float v8f;

#define BATCH   512
#define DIM     128
#define NCLS    50000
#define NTW     64            // classes per workgroup
#define ALPHA_C 32.0f
#define MRG_C   0.1f
#define STATW_C 0.01f

// workspace layout (float offsets)
#define WS_XN    0                       // 512*128 normalized X
#define WS_ADJ   (BATCH * DIM)           // 50048 adj values (zero padded)
#define WS_ADJ_N 50048
#define WS_NV    (WS_ADJ + WS_ADJ_N)     // num_valid (float)

// ---------------------------------------------------------------- init
__global__ void init_ws_kernel(float* ws, float* out) {
    int i = blockIdx.x * blockDim.x + threadIdx.x;
    if (i < WS_ADJ_N + 1) ws[WS_ADJ + i] = 0.0f;   // adj[] and num_valid
    if (i == 0) out[0] = 0.0f;
}

// ---------------------------------------------------------------- X row l2-norm (1 wave per row)
__global__ void xnorm_kernel(const float* __restrict__ X, float* __restrict__ ws) {
    int row  = blockIdx.x;          // 512 blocks
    int lane = threadIdx.x;         // 32 threads (wave32)
    const float* xr = X + (size_t)row * DIM;
    float v[4];
    float s = 0.0f;
    #pragma unroll
    for (int i = 0; i < 4; i++) { v[i] = xr[lane + 32 * i]; s += v[i] * v[i]; }
    #pragma unroll
    for (int off = 16; off > 0; off >>= 1) s += __shfl_xor(s, off, 32);
    float inv = 1.0f / sqrtf(s + 1e-12f);
    float* xn = ws + WS_XN + (size_t)row * DIM;
    #pragma unroll
    for (int i = 0; i < 4; i++) xn[lane + 32 * i] = v[i] * inv;
}

// ---------------------------------------------------------------- per-class stats -> adj[], num_valid
__device__ __forceinline__ float block_reduce128(float v, float* red, int d) {
    red[d] = v; __syncthreads();
    #pragma unroll
    for (int off = 64; off > 0; off >>= 1) {
        if (d < off) red[d] += red[d + off];
        __syncthreads();
    }
    float r = red[0]; __syncthreads();
    return r;
}

__global__ void stats_kernel(const int* __restrict__ T, const float* __restrict__ proxies,
                             const float* __restrict__ centers, float* __restrict__ ws) {
    __shared__ int   Ts[BATCH];
    __shared__ float red[DIM];
    int b = blockIdx.x;        // 512 blocks (one per sample)
    int d = threadIdx.x;       // 128 threads (one per feature dim)
    for (int i = d; i < BATCH; i += DIM) Ts[i] = T[i];
    __syncthreads();
    int t = Ts[b];
    bool first = true;
    for (int j = 0; j < b; j++) if (Ts[j] == t) { first = false; break; }
    if (!first) return;        // uniform across block

    const float* xn = ws + WS_XN;
    float s = 0.0f, q = 0.0f; int cnt = 0;
    for (int j = 0; j < BATCH; j++) {
        if (Ts[j] == t) { float x = xn[(size_t)j * DIM + d]; s += x; q += x * x; cnt++; }
    }
    float fc   = (float)cnt;                    // >= 1 here
    float mean = s / fc;
    float var  = q / fc - mean * mean;
    var = fminf(fmaxf(var, 1e-6f), 10.0f);

    float var_sum = block_reduce128(var, red, d);
    float vw = 1.0f / (1.0f + var_sum / (float)DIM);

    float p  = proxies[(size_t)t * DIM + d];
    float cc = centers[(size_t)t * DIM + d];
    float S  = block_reduce128(p * p,  red, d);
    float Cs = block_reduce128(cc * cc, red, d);
    float Dt = block_reduce128(p * cc,  red, d);

    float pn_scale = 1.0f / sqrtf(S + 1e-12f);      // Pn = p * pn_scale
    float pnorm = sqrtf(S) * pn_scale;              // = sqrt(sum Pn^2)
    float cnorm = sqrtf(Cs);
    float dot   = Dt * pn_scale;
    float csim  = dot / (fmaxf(pnorm, 1e-8f) * fmaxf(cnorm, 1e-8f));

    if (d == 0) {
        ws[WS_ADJ + t] = csim * vw * 0.15f;
        atomicAdd(ws + WS_NV, 1.0f);
    }
}

// ---------------------------------------------------------------- center_reg = mean|centers| * 0.01
__global__ void reg_kernel(const float* __restrict__ centers, float* __restrict__ out) {
    __shared__ float red[256];
    const size_t n = (size_t)NCLS * DIM;
    float acc = 0.0f;
    for (size_t i = (size_t)blockIdx.x * blockDim.x + threadIdx.x; i < n;
         i += (size_t)gridDim.x * blockDim.x)
        acc += fabsf(centers[i]);
    red[threadIdx.x] = acc; __syncthreads();
    #pragma unroll
    for (int off = 128; off > 0; off >>= 1) {
        if ((int)threadIdx.x < off) red[threadIdx.x] += red[threadIdx.x + off];
        __syncthreads();
    }
    if (threadIdx.x == 0) atomicAdd(out, STATW_C * red[0] / (float)n);
}

// ---------------------------------------------------------------- fused WMMA GEMM + proxy-anchor terms
__global__ void __launch_bounds__(256, 2)
pa_gemm_kernel(const int* __restrict__ T, const float* __restrict__ proxies,
               const float* __restrict__ ws, float* __restrict__ out) {
    __shared__ float Pn[NTW][DIM + 4];      // +4 pad: B-frag ds_load_b64 hits all 64 banks
    __shared__ float rnorm[NTW];
    __shared__ int   Tsh[BATCH];
    __shared__ float posAcc[NTW], negAcc[NTW];
    __shared__ float redc[NTW];

    const int tid = threadIdx.x;
    const int c0  = blockIdx.x * NTW;

    // stage raw proxy tile (zero-pad past NCLS)
    for (int i = tid; i < NTW * DIM; i += 256) {
        int r = i / DIM, k = i % DIM;
        int cls = c0 + r;
        Pn[r][k] = (cls < NCLS) ? proxies[(size_t)cls * DIM + k] : 0.0f;
    }
    for (int i = tid; i < BATCH; i += 256) Tsh[i] = T[i];
    if (tid < NTW) { posAcc[tid] = 0.0f; negAcc[tid] = 0.0f; }
    __syncthreads();

    if (tid < NTW) {
        float s = 0.0f;
        for (int k = 0; k < DIM; k++) { float p = Pn[tid][k]; s += p * p; }
        rnorm[tid] = 1.0f / sqrtf(s + 1e-12f);
    }
    __syncthreads();
    for (int i = tid; i < NTW * DIM; i += 256) {
        int r = i / DIM, k = i % DIM;
        Pn[r][k] *= rnorm[r];
    }
    __syncthreads();

    const int wave = tid >> 5, lane = tid & 31;
    const int half = lane >> 4, lq = lane & 15;

    const float invNV = 1.0f / ws[WS_NV];
    const float* Xn  = ws + WS_XN;
    const float* adj = ws + WS_ADJ;

    float adjv[4];
    #pragma unroll
    for (int nt = 0; nt < 4; nt++) adjv[nt] = adj[c0 + nt * 16 + lq];  // zero in pad region

    // B fragment base pointers: lane supplies N = lq, K = j + 2*half per VGPR j
    const float* bbase[4];
    #pragma unroll
    for (int nt = 0; nt < 4; nt++) bbase[nt] = &Pn[nt * 16 + lq][2 * half];

    // each wave covers 4 M-tiles (8 waves * 4 * 16 = 512 rows)
    for (int mt = 0; mt < 4; mt++) {
        const int m0 = (wave * 4 + mt) * 16;
        const float* arow = Xn + (size_t)(m0 + lq) * DIM + 2 * half;  // A: M=lq, K=j+2*half

        v8f acc[4] = {};
        for (int k = 0; k < DIM; k += 4) {
            v2f a = *(const v2f*)(arow + k);          // 8B-aligned (k even, 2*half even)
            #pragma unroll
            for (int nt = 0; nt < 4; nt++) {
                v2f bfrag = *(const v2f*)(bbase[nt] + k);
                acc[nt] = __builtin_amdgcn_wmma_f32_16x16x4_f32(
                    /*neg_a=*/false, a, /*neg_b=*/false, bfrag,
                    /*c_mod=*/(short)0, acc[nt], /*reuse_a=*/false, /*reuse_b=*/false);
            }
        }

        // T labels for this M-tile (independent of nt -> hoisted)
        int tvals[8];
        #pragma unroll
        for (int j = 0; j < 8; j++) tvals[j] = Tsh[m0 + j + 8 * half];

        // post-process: lane holds N=lq column for M = m0 + j + 8*half, j=0..7
        #pragma unroll
        for (int nt = 0; nt < 4; nt++) {
            const int nloc = nt * 16 + lq;
            const int cls  = c0 + nloc;
            float poss = 0.0f, negs = 0.0f;
            #pragma unroll
            for (int j = 0; j < 8; j++) {
                float comb = acc[nt][j] + adjv[nt];
                bool  isp  = (tvals[j] == cls);
                // single v_exp per element, branchless via cndmask on the argument
                float arg  = isp ? (-ALPHA_C * (comb - MRG_C))
                                 : ( ALPHA_C * (comb + MRG_C));
                float e    = __expf(arg);
                poss += isp ? e : 0.0f;
                negs += isp ? 0.0f : e;
            }
            // fold the two lane-halves (same N, disjoint M)
            poss += __shfl_xor(poss, 16, 32);
            negs += __shfl_xor(negs, 16, 32);
            if (half == 0) {
                atomicAdd(&posAcc[nloc], poss);   // ds_add_f32
                atomicAdd(&negAcc[nloc], negs);
            }
        }
    }
    __syncthreads();

    // per-class terms, block-reduced -> one global atomic per workgroup
    float contrib = 0.0f;
    if (tid < NTW) {
        int cls = c0 + tid;
        if (cls < NCLS) {
            contrib = log1pf(posAcc[tid]) * invNV
                    + log1pf(negAcc[tid]) * (1.0f / (float)NCLS);
        }
        redc[tid] = contrib;
    }
    __syncthreads();
    if (tid < NTW) {
        #pragma unroll
        for (int off = 32; off > 0; off >>= 1) {
            if (tid < off) redc[tid] += redc[tid + off];
            __syncthreads();
        }
        if (tid == 0) atomicAdd(out, redc[0]);
    }
}

// ---------------------------------------------------------------- launch
extern "C" void kernel_launch(void* const* d_in, const int* in_sizes, int n_in,
                              void* d_out, int out_size, void* d_ws, size_t ws_size,
                              hipStream_t stream) {
    const float* X       = (const float*)d_in[0];
    const int*   T       = (const int*)  d_in[1];
    const float* proxies = (const float*)d_in[2];
    const float* centers = (const float*)d_in[3];
    float* out = (float*)d_out;
    float* ws  = (float*)d_ws;

    init_ws_kernel<<<(WS_ADJ_N + 1 + 255) / 256, 256, 0, stream>>>(ws, out);
    xnorm_kernel  <<<BATCH, 32, 0, stream>>>(X, ws);
    stats_kernel  <<<BATCH, DIM, 0, stream>>>(T, proxies, centers, ws);
    reg_kernel    <<<512, 256, 0, stream>>>(centers, out);
    pa_gemm_kernel<<<(NCLS + NTW - 1) / NTW, 256, 0, stream>>>(T, proxies, ws, out);
}